// MomentumGD_51462298141306
// MI455X (gfx1250) — compile-verified
//
#include <hip/hip_runtime.h>

// Problem constants (match reference: B=8, T=4096, D=1024, fp32)
#define BB   8
#define TT   4096
#define DD   1024
#define DV   (DD / 4)    // 256 float4 groups along D
#define CL   64          // chunk length along T
#define NC   (TT / CL)   // 64 chunks
#define PF   16          // prefetch distance (rows of 4KB => 8KB/wave ahead)

typedef float v4f __attribute__((ext_vector_type(4)));
typedef int   v4i __attribute__((ext_vector_type(4)));

#define AS_GLOBAL __attribute__((address_space(1)))
#define AS_LDS    __attribute__((address_space(3)))

// ---- gfx1250 async-LDS path feature detection (compile-safe) --------------
#if defined(__has_builtin)
#  if __has_builtin(__builtin_amdgcn_global_load_async_to_lds_b128)
#    define HAVE_ASYNC_LDS 1
#  endif
#  if __has_builtin(__builtin_amdgcn_s_wait_asynccnt)
#    define HAVE_WAIT_ASYNC 1
#  endif
#endif

template <int N>
__device__ __forceinline__ void wait_asynccnt() {
#if defined(HAVE_WAIT_ASYNC)
    __builtin_amdgcn_s_wait_asynccnt(N);
#else
    asm volatile("s_wait_asynccnt %0" :: "n"(N) : "memory");
#endif
    asm volatile("" ::: "memory");   // keep ds_loads below the wait
}

// ---------------------------------------------------------------------------
// Pass 1 (direct-load variant): per (b, chunk, dv) compute local chunk scan
// with m0 = 0. Writes s_c to ws. ws layout: [b][dv][c] as v4f.
// ---------------------------------------------------------------------------
__global__ void __launch_bounds__(256)
mgd_pass1_chunkscan(const float* __restrict__ g,
                    const float* __restrict__ beta,
                    float* __restrict__ ws) {
    int gt   = blockIdx.x * blockDim.x + threadIdx.x;   // [0, BB*NC*DV)
    int dv   = gt & (DV - 1);
    int rest = gt >> 8;          // / DV
    int c    = rest & (NC - 1);
    int b    = rest >> 6;        // / NC

    const v4f* __restrict__ bp = (const v4f*)beta;
    v4f b4  = bp[dv];
    v4f omb = 1.0f - b4;

    const v4f* __restrict__ gp =
        (const v4f*)g + (size_t)(b * TT + c * CL) * DV + dv;

    v4f m = (v4f)(0.0f);
#pragma unroll 8
    for (int i = 0; i < CL; ++i) {
        __builtin_prefetch((const void*)(gp + (size_t)(i + PF) * DV), 0, 3);
        v4f x = gp[(size_t)i * DV];
        m = b4 * m + omb * x;
    }

    v4f* __restrict__ wp = (v4f*)ws;
    wp[(size_t)(b * DV + dv) * NC + c] = m;
}

#if defined(HAVE_ASYNC_LDS)
// ---------------------------------------------------------------------------
// Pass 1 (async-LDS variant): grads staged into LDS with
// global_load_async_to_lds_b128 in a 4-deep pipeline of 32KB tiles
// (128KB LDS/block => 2 blocks per 320KB WGP). Each lane copies exactly the
// 16B it later consumes, and per-wave async loads complete in order, so the
// only sync needed is s_wait_asynccnt — no workgroup barriers.
// ---------------------------------------------------------------------------
#define TROWS 8                   // T-rows per tile
#define NTILE (CL / TROWS)        // 8 tiles per chunk
#define NBUF  4                   // pipeline depth; (NBUF-1)*TROWS=24 <= 63
#define ROWB  (DD * 4)            // 4096 bytes per T-row
#define TILEB (TROWS * ROWB)      // 32 KB per tile

__global__ void __launch_bounds__(256)
mgd_pass1_async(const float* __restrict__ g,
                const float* __restrict__ beta,
                float* __restrict__ ws) {
    extern __shared__ char smem[];          // NBUF * TILEB = 128 KB dynamic
    const int dv = threadIdx.x;             // 0..255
    const int bc = blockIdx.x;              // [0, BB*NC)
    const int c  = bc & (NC - 1);
    const int b  = bc >> 6;

    v4f b4  = ((const v4f*)beta)[dv];
    v4f omb = 1.0f - b4;

    const char* gbase =
        (const char*)g + ((size_t)(b * TT + c * CL) * DD + (size_t)dv * 4) * 4;
    char* lbase = smem + (size_t)dv * 16;

    auto issue_tile = [&](int t) {
        const char* gs = gbase + (size_t)t * TILEB;
        char*       ls = lbase + (size_t)(t & (NBUF - 1)) * TILEB;
#pragma unroll
        for (int r = 0; r < TROWS; ++r) {
            __builtin_amdgcn_global_load_async_to_lds_b128(
                (AS_GLOBAL v4i*)(gs + r * ROWB),
                (AS_LDS    v4i*)(ls + r * ROWB),
                /*imm offset*/ 0, /*cpol*/ 0);
        }
    };

#pragma unroll
    for (int p = 0; p < NBUF - 1; ++p) issue_tile(p);

    v4f m = (v4f)(0.0f);
#pragma unroll
    for (int t = 0; t < NTILE; ++t) {
        if (t + NBUF - 1 < NTILE) issue_tile(t + NBUF - 1);
        // tiles still in flight after tile t has landed:
        const int pend = ((NTILE - 1 - t) < (NBUF - 1) ? (NTILE - 1 - t)
                                                       : (NBUF - 1)) * TROWS;
        if      (pend >= 3 * TROWS) wait_asynccnt<3 * TROWS>();
        else if (pend == 2 * TROWS) wait_asynccnt<2 * TROWS>();
        else if (pend == 1 * TROWS) wait_asynccnt<1 * TROWS>();
        else                        wait_asynccnt<0>();

        const v4f* ls = (const v4f*)(smem + (size_t)(t & (NBUF - 1)) * TILEB) + dv;
#pragma unroll
        for (int r = 0; r < TROWS; ++r) {
            v4f x = ls[(size_t)r * DV];     // ds_load_b128
            m = b4 * m + omb * x;
        }
    }

    ((v4f*)ws)[(size_t)(b * DV + dv) * NC + c] = m;
}
#endif // HAVE_ASYNC_LDS

// ---------------------------------------------------------------------------
// Pass 2: per chain (b, dv), turn chunk sums s_c into carry-ins:
//   carry_in[0] = 0 ; carry_in[c] = beta^CL * carry_in[c-1] + s[c-1]
// ---------------------------------------------------------------------------
__global__ void __launch_bounds__(256)
mgd_pass2_combine(const float* __restrict__ beta,
                  float* __restrict__ ws) {
    int gt = blockIdx.x * blockDim.x + threadIdx.x;     // [0, BB*DV)
    int dv = gt & (DV - 1);
    int b  = gt >> 8;

    const v4f* __restrict__ bp = (const v4f*)beta;
    v4f bL = bp[dv];
#pragma unroll
    for (int k = 0; k < 6; ++k) bL = bL * bL;           // beta^64

    v4f* __restrict__ p = (v4f*)ws + (size_t)(b * DV + dv) * NC;
    v4f cur = (v4f)(0.0f);
#pragma unroll 4
    for (int c = 0; c < NC; ++c) {
        v4f s = p[c];
        p[c]  = cur;                 // carry-in for chunk c
        cur   = bL * cur + s;
    }
}

// ---------------------------------------------------------------------------
// Pass 3: per (b, chunk, dv), rescan chunk from exact carry-in, emit outputs.
// Second grad read hits the 192MB L2 (filled by pass 1); loads marked
// non-temporal (last use); stores non-temporal to avoid evicting grads.
// ---------------------------------------------------------------------------
__global__ void __launch_bounds__(256)
mgd_pass3_apply(const float* __restrict__ g,
                const float* __restrict__ beta,
                const float* __restrict__ eta,
                const float* __restrict__ ws,
                float* __restrict__ out) {
    int gt   = blockIdx.x * blockDim.x + threadIdx.x;
    int dv   = gt & (DV - 1);
    int rest = gt >> 8;
    int c    = rest & (NC - 1);
    int b    = rest >> 6;

    const v4f* __restrict__ bp = (const v4f*)beta;
    const v4f* __restrict__ ep = (const v4f*)eta;
    v4f b4  = bp[dv];
    v4f omb = 1.0f - b4;
    v4f e4  = ep[dv];

    const v4f* __restrict__ wp = (const v4f*)ws;
    v4f m = wp[(size_t)(b * DV + dv) * NC + c];

    size_t base = (size_t)(b * TT + c * CL) * DV + dv;
    const v4f* __restrict__ gp = (const v4f*)g + base;
    v4f* __restrict__ op       = (v4f*)out + base;

#pragma unroll 8
    for (int i = 0; i < CL; ++i) {
        __builtin_prefetch((const void*)(gp + (size_t)(i + PF) * DV), 0, 3);
        v4f x = __builtin_nontemporal_load(gp + (size_t)i * DV);
        m = b4 * m + omb * x;
        v4f o = -e4 * m;
        __builtin_nontemporal_store(o, op + (size_t)i * DV);
    }
}

// ---------------------------------------------------------------------------
// Fallback single-pass kernel (only if workspace is too small).
// ---------------------------------------------------------------------------
__global__ void __launch_bounds__(256)
mgd_singlepass(const float* __restrict__ g,
               const float* __restrict__ beta,
               const float* __restrict__ eta,
               float* __restrict__ out) {
    int gt = blockIdx.x * blockDim.x + threadIdx.x;     // [0, BB*DV)
    int dv = gt & (DV - 1);
    int b  = gt >> 8;

    const v4f* __restrict__ bp = (const v4f*)beta;
    const v4f* __restrict__ ep = (const v4f*)eta;
    v4f b4  = bp[dv];
    v4f omb = 1.0f - b4;
    v4f e4  = ep[dv];

    size_t base = (size_t)(b * TT) * DV + dv;
    const v4f* __restrict__ gp = (const v4f*)g + base;
    v4f* __restrict__ op       = (v4f*)out + base;

    v4f m = (v4f)(0.0f);
#pragma unroll 4
    for (int t = 0; t < TT; ++t) {
        __builtin_prefetch((const void*)(gp + (size_t)(t + PF) * DV), 0, 3);
        v4f x = gp[(size_t)t * DV];
        m = b4 * m + omb * x;
        v4f o = -e4 * m;
        __builtin_nontemporal_store(o, op + (size_t)t * DV);
    }
}

extern "C" void kernel_launch(void* const* d_in, const int* in_sizes, int n_in,
                              void* d_out, int out_size, void* d_ws, size_t ws_size,
                              hipStream_t stream) {
    const float* grads = (const float*)d_in[0];   // [B, T, D] fp32
    const float* eta   = (const float*)d_in[1];   // [D] fp32
    const float* beta  = (const float*)d_in[2];   // [D] fp32
    float*       out   = (float*)d_out;           // [B, T, D] fp32
    float*       ws    = (float*)d_ws;

    const size_t ws_needed = (size_t)BB * DV * NC * sizeof(v4f);  // 2 MiB

    if (ws_size >= ws_needed) {
        const int n_chunk_threads = BB * NC * DV;   // 131072
        const int n_chain_threads = BB * DV;        // 2048

#if defined(HAVE_ASYNC_LDS)
        mgd_pass1_async<<<BB * NC, 256, NBUF * TILEB, stream>>>(
            grads, beta, ws);
#else
        mgd_pass1_chunkscan<<<n_chunk_threads / 256, 256, 0, stream>>>(
            grads, beta, ws);
#endif
        mgd_pass2_combine<<<n_chain_threads / 256, 256, 0, stream>>>(
            beta, ws);
        mgd_pass3_apply<<<n_chunk_threads / 256, 256, 0, stream>>>(
            grads, beta, eta, ws, out);
    } else {
        const int n_chain_threads = BB * DV;        // 2048
        mgd_singlepass<<<n_chain_threads / 256, 256, 0, stream>>>(
            grads, beta, eta, out);
    }
}